// InteractionLayer_76897094467615
// MI455X (gfx1250) — compile-verified
//
#include <hip/hip_runtime.h>

typedef __attribute__((ext_vector_type(2))) float v2f;
typedef __attribute__((ext_vector_type(4))) float v4f;
typedef __attribute__((ext_vector_type(8))) float v8f;

// Edge kernel: 16 edges per wave-iteration on the WMMA unit, transposed mapping:
//   A (16x4 f32): row M=0 = weight chunk (held by lanes 0 and 16, all other rows 0)
//   B (4x16 f32): column N=sub = edge sub's features; lane (sub,half) holds
//                 f[sub][2*half], f[sub][2*half+1]  -> one b64/two b32 loads per lane
//   D (16x16 f32): row M=0 = messages: D VGPR0, lane n (n<16) = msg of edge n.
// No cross-lane redistribution needed: msg and dst[e] are already in the same lane.
__global__ __launch_bounds__(256) void edge_msgs_wmma(
    const float* __restrict__ z, const float* __restrict__ r,
    const float* __restrict__ rhat, const float* __restrict__ W,
    const float* __restrict__ b, const int* __restrict__ src,
    const int* __restrict__ dst, float* __restrict__ agg, int E, int layer) {
  const int lane = threadIdx.x & 31;
  const int sub  = lane & 15;       // edge slot within the 16-edge chunk (N index)
  const int half = lane >> 4;       // which K-pair of the chunk this lane feeds
  const int wave   = (blockIdx.x * blockDim.x + threadIdx.x) >> 5;
  const int nwaves = (gridDim.x * blockDim.x) >> 5;

  const float* w   = W + 12 * layer;
  const float bias = b[layer];

  // A operands (hoisted: constant per layer). Row M=0 of the 16x4 A matrix lives in
  // lane 0 (K = 0,1) and lane 16 (K = 2,3); every other lane contributes zero rows.
  const bool wl = (sub == 0);
  v2f A0 = { wl ? w[0 + 2 * half] : 0.0f, wl ? w[1 + 2 * half] : 0.0f };
  v2f A1 = { wl ? w[4 + 2 * half] : 0.0f, wl ? w[5 + 2 * half] : 0.0f };
  v2f A2 = { wl ? w[8 + 2 * half] : 0.0f, wl ? w[9 + 2 * half] : 0.0f };

  const int nch = (E + 15) >> 4;
  for (int ch = wave; ch < nch; ch += nwaves) {
    const int e  = (ch << 4) + sub;
    const int ec = e < E ? e : (E - 1);      // clamp tail; atomic is guarded below
    const int si = src[ec];
    const int di = dst[ec];

    // Prefetch next chunk of the streaming (L2-resident) edge arrays.
    // Speculative prefetch: OOB translation failures are silently dropped.
    const int en = ec + (nwaves << 4);
    __builtin_prefetch(src + en, 0, 0);
    __builtin_prefetch(dst + en, 0, 0);
    __builtin_prefetch(r + en, 0, 0);
    __builtin_prefetch(rhat + 3 * en, 0, 0);

    // B chunks: z[src] (K0..3), z[dst] (K4..7), [r, rhat0..2] (K8..11).
    v2f B0 = *reinterpret_cast<const v2f*>(z + 4 * si + 2 * half);
    v2f B1 = *reinterpret_cast<const v2f*>(z + 4 * di + 2 * half);
    // Branchless per-half feature loads: select addresses, not control flow.
    const float* p0 = half ? (rhat + 3 * ec + 1) : (r + ec);
    const float* p1 = half ? (rhat + 3 * ec + 2) : (rhat + 3 * ec);
    v2f B2 = { *p0, *p1 };

    v8f C = {0.0f, 0.0f, 0.0f, 0.0f, 0.0f, 0.0f, 0.0f, 0.0f};
    C = __builtin_amdgcn_wmma_f32_16x16x4_f32(false, A0, false, B0, (short)0, C, false, false);
    C = __builtin_amdgcn_wmma_f32_16x16x4_f32(false, A1, false, B1, (short)0, C, false, false);
    C = __builtin_amdgcn_wmma_f32_16x16x4_f32(false, A2, false, B2, (short)0, C, false, false);

    // D VGPR0, lanes 0..15 = messages for edges ch*16 + 0..15, already lane-aligned.
    if (lane < 16 && e < E)
      atomicAdd(agg + di, C[0] + bias);      // global_atomic_add_f32, L2-resident agg
  }
}

__global__ __launch_bounds__(256) void node_update(float* __restrict__ zw,
                                                   float* __restrict__ agg, int N) {
  int n = blockIdx.x * blockDim.x + threadIdx.x;
  if (n < N) {
    float a = agg[n];
    agg[n] = 0.0f;                            // ready for next layer / next call
    v4f zv = *reinterpret_cast<v4f*>(zw + 4 * n);
    zv.x += a; zv.y += a; zv.z += a; zv.w += a;
    *reinterpret_cast<v4f*>(zw + 4 * n) = zv;
  }
}

__global__ __launch_bounds__(256) void init_state(const float* __restrict__ z,
                                                  float* __restrict__ zw,
                                                  float* __restrict__ x,
                                                  float* __restrict__ agg, int N) {
  int n = blockIdx.x * blockDim.x + threadIdx.x;
  if (n < N) {
    v4f zv = *reinterpret_cast<const v4f*>(z + 4 * n);
    *reinterpret_cast<v4f*>(zw + 4 * n) = zv;  // working copy (also output slot 0)
    *reinterpret_cast<v4f*>(x + 4 * n)  = zv;  // x = z.copy() (output slot 1)
    agg[n] = 0.0f;
  }
}

extern "C" void kernel_launch(void* const* d_in, const int* in_sizes, int n_in,
                              void* d_out, int out_size, void* d_ws, size_t ws_size,
                              hipStream_t stream) {
  const float* z    = (const float*)d_in[0];
  const float* r    = (const float*)d_in[1];
  const float* rhat = (const float*)d_in[2];
  const float* W    = (const float*)d_in[3];
  const float* b    = (const float*)d_in[4];
  const int*   src  = (const int*)d_in[5];
  const int*   dst  = (const int*)d_in[6];

  const int N = in_sizes[0] / 4;   // z is [N,4]
  const int E = in_sizes[5];       // src is [E]
  const int L = in_sizes[3] / 12;  // W is [L,1,12]

  float* out = (float*)d_out;
  float* zw  = out;                      // z_final written here (first N*4 floats)
  float* x   = out + (size_t)N * 4;      // original z copy (second N*4 floats)
  float* agg = (float*)d_ws;             // N floats of scratch (0.8 MB, L2-resident)

  const int nb = (N + 255) / 256;
  init_state<<<nb, 256, 0, stream>>>(z, zw, x, agg, N);
  for (int layer = 0; layer < L; ++layer) {
    edge_msgs_wmma<<<2048, 256, 0, stream>>>(zw, r, rhat, W, b, src, dst, agg, E, layer);
    node_update<<<nb, 256, 0, stream>>>(zw, agg, N);
  }
}